// Memory_60352880444052
// MI455X (gfx1250) — compile-verified
//
#include <hip/hip_runtime.h>
#include <hip/hip_bf16.h>
#include <math.h>

#define DNC_B 8
#define DNC_N 4096
#define DNC_W 64
#define DNC_R 4
#define DNC_EPS 1e-8f

typedef __attribute__((ext_vector_type(2))) float v2f;
typedef __attribute__((ext_vector_type(4))) float v4f;
typedef __attribute__((ext_vector_type(8))) float v8f;

#ifndef __has_builtin
#define __has_builtin(x) 0
#endif
#if defined(__HIP_DEVICE_COMPILE__) && defined(__gfx1250__) && \
    __has_builtin(__builtin_amdgcn_wmma_f32_16x16x4_f32)
#define HAVE_WMMA_F32 1
#else
#define HAVE_WMMA_F32 0
#endif

// ---------------------------------------------------------------------------
// K0: per-(b,r) and per-b key norms (tiny)
// ---------------------------------------------------------------------------
__global__ void norms_kernel(const float* __restrict__ wkey,
                             const float* __restrict__ rkeys,
                             float* __restrict__ normk,
                             float* __restrict__ normrk) {
  const int t = threadIdx.x;
  if (t < DNC_B * DNC_R) {
    const int b = t >> 2, r = t & 3;
    float s = 0.f;
    for (int w = 0; w < DNC_W; ++w) {
      float v = rkeys[(b * DNC_W + w) * DNC_R + r];
      s += v * v;
    }
    normrk[t] = sqrtf(s);
  } else if (t < DNC_B * DNC_R + DNC_B) {
    const int b = t - DNC_B * DNC_R;
    float s = 0.f;
    for (int w = 0; w < DNC_W; ++w) {
      float v = wkey[b * DNC_W + w];
      s += v * v;
    }
    normk[b] = sqrtf(s);
  }
}

// ---------------------------------------------------------------------------
// K1: new memory usage + scaled write-content similarity logits
// ---------------------------------------------------------------------------
__global__ __launch_bounds__(256) void usage_ssim_kernel(
    const float* __restrict__ mem, const float* __restrict__ wkey,
    const float* __restrict__ musage, const float* __restrict__ wwt,
    const float* __restrict__ rdw, const float* __restrict__ fgates,
    const float* __restrict__ wstr, const float* __restrict__ normk,
    float* __restrict__ usage_out, float* __restrict__ ssim_out) {
  const int idx = blockIdx.x * 256 + threadIdx.x;
  if (idx >= DNC_B * DNC_N) return;
  const int b = idx / DNC_N;
  float u = musage[idx], w = wwt[idx];
  float uw = u + w - u * w;
  float ur = 1.f;
#pragma unroll
  for (int r = 0; r < DNC_R; ++r)
    ur *= (1.f - rdw[(size_t)idx * DNC_R + r] * fgates[b * DNC_R + r]);
  usage_out[idx] = uw * ur;

  const float* mrow = mem + (size_t)idx * DNC_W;
  const float* wk = wkey + b * DNC_W;
  float dot = 0.f, nm = 0.f;
#pragma unroll 8
  for (int k = 0; k < DNC_W; ++k) {
    float m = mrow[k];
    dot = fmaf(m, wk[k], dot);
    nm = fmaf(m, m, nm);
  }
  ssim_out[idx] = wstr[b] * dot / (normk[b] * sqrtf(nm) + DNC_EPS);
}

// ---------------------------------------------------------------------------
// K2: softmax over N for the write content weighting (one block per batch)
// ---------------------------------------------------------------------------
__global__ __launch_bounds__(256) void softmax_cbw_kernel(
    const float* __restrict__ ssim, float* __restrict__ cbw) {
  const int b = blockIdx.x, t = threadIdx.x;
  __shared__ float red[256];
  float m = -3.0e38f;
  for (int n = t; n < DNC_N; n += 256) m = fmaxf(m, ssim[b * DNC_N + n]);
  red[t] = m;
  __syncthreads();
  for (int s = 128; s > 0; s >>= 1) {
    if (t < s) red[t] = fmaxf(red[t], red[t + s]);
    __syncthreads();
  }
  m = red[0];
  __syncthreads();
  float sum = 0.f;
  for (int n = t; n < DNC_N; n += 256) sum += expf(ssim[b * DNC_N + n] - m);
  red[t] = sum;
  __syncthreads();
  for (int s = 128; s > 0; s >>= 1) {
    if (t < s) red[t] += red[t + s];
    __syncthreads();
  }
  const float inv = 1.f / red[0];
  for (int n = t; n < DNC_N; n += 256)
    cbw[b * DNC_N + n] = expf(ssim[b * DNC_N + n] - m) * inv;
}

// ---------------------------------------------------------------------------
// K3: per-batch bitonic argsort of usage + prefix-product allocation weights
//     and final write weighting ww. 48 KB LDS (fine inside 320KB/WGP).
// ---------------------------------------------------------------------------
__global__ __launch_bounds__(1024) void sort_alloc_ww_kernel(
    const float* __restrict__ usage, const float* __restrict__ cbw,
    const float* __restrict__ agate, const float* __restrict__ wgate,
    float* __restrict__ wwout) {
  const int b = blockIdx.x, t = threadIdx.x;
  __shared__ float sv[DNC_N];
  __shared__ int si[DNC_N];
  __shared__ float pp[DNC_N];

  for (int i = t; i < DNC_N; i += 1024) {
    sv[i] = usage[b * DNC_N + i];
    si[i] = i;
  }
  __syncthreads();

  // bitonic sort, ascending
  for (int k = 2; k <= DNC_N; k <<= 1) {
    for (int j = k >> 1; j > 0; j >>= 1) {
      for (int i = t; i < DNC_N; i += 1024) {
        const int ixj = i ^ j;
        if (ixj > i) {
          const bool up = ((i & k) == 0);
          float a = sv[i], c = sv[ixj];
          if ((a > c) == up) {
            sv[i] = c; sv[ixj] = a;
            int ia = si[i]; si[i] = si[ixj]; si[ixj] = ia;
          }
        }
      }
      __syncthreads();
    }
  }

  // inclusive multiplicative scan (Hillis-Steele), 4 elements / thread
  for (int i = t; i < DNC_N; i += 1024) pp[i] = sv[i];
  __syncthreads();
  for (int off = 1; off < DNC_N; off <<= 1) {
    float tmp[4];
#pragma unroll
    for (int c = 0; c < 4; ++c) {
      const int i = t + c * 1024;
      tmp[c] = (i >= off) ? pp[i - off] * pp[i] : pp[i];
    }
    __syncthreads();
#pragma unroll
    for (int c = 0; c < 4; ++c) pp[t + c * 1024] = tmp[c];
    __syncthreads();
  }

  const float ag = agate[b], wg = wgate[b];
  for (int i = t; i < DNC_N; i += 1024) {
    const float cumex = (i == 0) ? 1.f : pp[i - 1];    // exclusive prefix prod
    const float alloc = (1.f - sv[i]) * cumex;
    const int n = si[i];                               // scatter to order[i]
    wwout[b * DNC_N + n] =
        wg * (ag * alloc + (1.f - ag) * cbw[b * DNC_N + n]);
  }
}

// ---------------------------------------------------------------------------
// K3b: pack per-m operands for the link kernels.
//   pkA[m] = (1-ww[m], p[m])   (fwd A transform)
//   pkB[m] = (1-ww[m], ww[m])  (bwd A transform)
//   bpack[pair][col] = (rwpad[2p][col], rwpad[2p+1][col]), cols 4..15 zero
// ---------------------------------------------------------------------------
__global__ __launch_bounds__(256) void prep_kernel(
    const float* __restrict__ ww, const float* __restrict__ prec,
    const float* __restrict__ rdw, float* __restrict__ pkA,
    float* __restrict__ pkB, float* __restrict__ bpack) {
  const int idx = blockIdx.x * 256 + threadIdx.x;
  if (idx >= DNC_B * DNC_N) return;
  const float w = ww[idx];
  const float omw = 1.f - w;
  const float p = prec[idx];
  pkA[2 * idx + 0] = omw;
  pkA[2 * idx + 1] = p;
  pkB[2 * idx + 0] = omw;
  pkB[2 * idx + 1] = w;
  if ((idx & 1) == 0) {
    const size_t pi = (size_t)(idx >> 1);
    float r0[4], r1[4];
#pragma unroll
    for (int r = 0; r < 4; ++r) {
      r0[r] = rdw[(size_t)idx * 4 + r];
      r1[r] = rdw[(size_t)(idx + 1) * 4 + r];
    }
#pragma unroll
    for (int c = 0; c < 16; ++c) {
      bpack[pi * 32 + 2 * c + 0] = (c < 4) ? r0[c] : 0.f;
      bpack[pi * 32 + 2 * c + 1] = (c < 4) ? r1[c] : 0.f;
    }
  }
}

// ---------------------------------------------------------------------------
// K4: memory erase/write + scaled read-content similarity logits
// ---------------------------------------------------------------------------
__global__ __launch_bounds__(256) void memupdate_kernel(
    const float* __restrict__ mem, const float* __restrict__ ww,
    const float* __restrict__ erase, const float* __restrict__ wvec,
    const float* __restrict__ rkeys, const float* __restrict__ rstr,
    const float* __restrict__ normrk, float* __restrict__ memn,
    float* __restrict__ ssr) {
  const int idx = blockIdx.x * 256 + threadIdx.x;
  if (idx >= DNC_B * DNC_N) return;
  const int b = idx / DNC_N;
  const float wwv = ww[idx];
  const float* mrow = mem + (size_t)idx * DNC_W;
  float* orow = memn + (size_t)idx * DNC_W;
  float nm = 0.f, d0 = 0.f, d1 = 0.f, d2 = 0.f, d3 = 0.f;
#pragma unroll 4
  for (int w = 0; w < DNC_W; ++w) {
    float mn = fmaf(wwv, wvec[b * DNC_W + w],
                    mrow[w] * (1.f - wwv * erase[b * DNC_W + w]));
    orow[w] = mn;
    nm = fmaf(mn, mn, nm);
    const float* rkp = rkeys + (size_t)(b * DNC_W + w) * DNC_R;
    d0 = fmaf(mn, rkp[0], d0);
    d1 = fmaf(mn, rkp[1], d1);
    d2 = fmaf(mn, rkp[2], d2);
    d3 = fmaf(mn, rkp[3], d3);
  }
  const float nmr = sqrtf(nm);
  ssr[(size_t)idx * 4 + 0] = rstr[b * 4 + 0] * d0 / (nmr * normrk[b * 4 + 0] + DNC_EPS);
  ssr[(size_t)idx * 4 + 1] = rstr[b * 4 + 1] * d1 / (nmr * normrk[b * 4 + 1] + DNC_EPS);
  ssr[(size_t)idx * 4 + 2] = rstr[b * 4 + 2] * d2 / (nmr * normrk[b * 4 + 2] + DNC_EPS);
  ssr[(size_t)idx * 4 + 3] = rstr[b * 4 + 3] * d3 / (nmr * normrk[b * 4 + 3] + DNC_EPS);
}

// ---------------------------------------------------------------------------
// K5: softmax over N per (b,r) for read content weighting
// ---------------------------------------------------------------------------
__global__ __launch_bounds__(256) void softmax_cbr_kernel(
    const float* __restrict__ ssr, float* __restrict__ cbr) {
  const int blk = blockIdx.x;
  const int b = blk >> 2, r = blk & 3, t = threadIdx.x;
  __shared__ float red[256];
  float m = -3.0e38f;
  for (int n = t; n < DNC_N; n += 256)
    m = fmaxf(m, ssr[((size_t)b * DNC_N + n) * 4 + r]);
  red[t] = m;
  __syncthreads();
  for (int s = 128; s > 0; s >>= 1) {
    if (t < s) red[t] = fmaxf(red[t], red[t + s]);
    __syncthreads();
  }
  m = red[0];
  __syncthreads();
  float sum = 0.f;
  for (int n = t; n < DNC_N; n += 256)
    sum += expf(ssr[((size_t)b * DNC_N + n) * 4 + r] - m);
  red[t] = sum;
  __syncthreads();
  for (int s = 128; s > 0; s >>= 1) {
    if (t < s) red[t] += red[t + s];
    __syncthreads();
  }
  const float inv = 1.f / red[0];
  for (int n = t; n < DNC_N; n += 256)
    cbr[((size_t)b * DNC_N + n) * 4 + r] =
        expf(ssr[((size_t)b * DNC_N + n) * 4 + r] - m) * inv;
}

// ---------------------------------------------------------------------------
// K6: fwd_w = L' x rw via V_WMMA_F32_16X16X4_F32, branchless inner loop.
// Diagonal handled only in the single 16-wide chunk mk == n0.
// ---------------------------------------------------------------------------
__global__ __launch_bounds__(256) void link_fwd_kernel(
    const float* __restrict__ L, const float* __restrict__ pkA,
    const float* __restrict__ bpack, const float* __restrict__ ww,
    float* __restrict__ fwdw) {
  const int b = blockIdx.y;
  const int lane = threadIdx.x & 31;
  const int wave = threadIdx.x >> 5;
  const int n0 = blockIdx.x * 128 + wave * 16;
  const int half = lane >> 4;
  const int l16 = lane & 15;
  const int col = l16;
  const int arow = n0 + l16;
  const float wwr = ww[b * DNC_N + arow];
  const float* Lrow = L + ((size_t)b * DNC_N + arow) * DNC_N;
  const float* pkAB = pkA + (size_t)b * DNC_N * 2;
  const float* bpB = bpack + (size_t)b * (DNC_N / 2) * 32;

#if HAVE_WMMA_F32
  v8f acc = {};
  auto step = [&](int k0, bool diag) {
    v2f lv = __builtin_nontemporal_load((const v2f*)(Lrow + k0));
    v4f wp = *(const v4f*)(pkAB + 2 * k0);  // {1-ww0, p0, 1-ww1, p1}
    v2f bv = *(const v2f*)(bpB + (size_t)(k0 >> 1) * 32 + 2 * col);
    v2f a;
    a.x = fmaf(wp.x - wwr, lv.x, wwr * wp.y);
    a.y = fmaf(wp.z - wwr, lv.y, wwr * wp.w);
    if (diag) {
      a.x = (arow == k0) ? 0.f : a.x;
      a.y = (arow == k0 + 1) ? 0.f : a.y;
    }
    acc = __builtin_amdgcn_wmma_f32_16x16x4_f32(false, a, false, bv, (short)0,
                                                acc, false, false);
  };
  for (int mk = 0; mk < n0; mk += 16) {
#pragma unroll
    for (int u = 0; u < 4; ++u) step(mk + u * 4 + half * 2, false);
  }
#pragma unroll
  for (int u = 0; u < 4; ++u) step(n0 + u * 4 + half * 2, true);
  for (int mk = n0 + 16; mk < DNC_N; mk += 16) {
#pragma unroll
    for (int u = 0; u < 4; ++u) step(mk + u * 4 + half * 2, false);
  }
  if (col < DNC_R) {
#pragma unroll
    for (int v = 0; v < 8; ++v) {
      const int m = v + half * 8;  // C rows: lanes16-31 hold M=8..15
      fwdw[((size_t)b * DNC_N + n0 + m) * DNC_R + col] = acc[v];
    }
  }
#else
  float acc[8];
#pragma unroll
  for (int v = 0; v < 8; ++v) acc[v] = 0.f;
  for (int mk = 0; mk < DNC_N; mk += 2) {
    const float* wp = pkAB + 2 * mk;
    const float* bp = bpB + (size_t)(mk >> 1) * 32 + 2 * col;
#pragma unroll
    for (int v = 0; v < 8; ++v) {
      const int rrow = n0 + half * 8 + v;
      const float wr = ww[b * DNC_N + rrow];
      const float* Lr = L + ((size_t)b * DNC_N + rrow) * DNC_N + mk;
      float a0 = fmaf(wp[0] - wr, Lr[0], wr * wp[1]);
      float a1 = fmaf(wp[2] - wr, Lr[1], wr * wp[3]);
      a0 = (rrow == mk) ? 0.f : a0;
      a1 = (rrow == mk + 1) ? 0.f : a1;
      acc[v] += a0 * bp[0] + a1 * bp[1];
    }
  }
  if (col < DNC_R)
#pragma unroll
    for (int v = 0; v < 8; ++v)
      fwdw[((size_t)b * DNC_N + n0 + half * 8 + v) * DNC_R + col] = acc[v];
#endif
}

// ---------------------------------------------------------------------------
// K7: bwd_w[n,r] = sum_m L'[m,n] * rw[m,r]; column-strip reads of L are
// contiguous 16-float rows -> coalesced half-wave loads.
// ---------------------------------------------------------------------------
__global__ __launch_bounds__(256) void link_bwd_kernel(
    const float* __restrict__ L, const float* __restrict__ pkB,
    const float* __restrict__ bpack, const float* __restrict__ ww,
    const float* __restrict__ prec, float* __restrict__ bwdw) {
  const int b = blockIdx.y;
  const int lane = threadIdx.x & 31;
  const int wave = threadIdx.x >> 5;
  const int n0 = blockIdx.x * 128 + wave * 16;
  const int half = lane >> 4;
  const int l16 = lane & 15;
  const int col = l16;
  const int an = n0 + l16;  // fixed L column == output row
  const float wwn = ww[b * DNC_N + an];
  const float pn = prec[b * DNC_N + an];
  const float* Lcol = L + (size_t)b * DNC_N * DNC_N + an;
  const float* pkBB = pkB + (size_t)b * DNC_N * 2;
  const float* bpB = bpack + (size_t)b * (DNC_N / 2) * 32;

#if HAVE_WMMA_F32
  v8f acc = {};
  auto step = [&](int k0, bool diag) {
    float l0 = __builtin_nontemporal_load(Lcol + (size_t)k0 * DNC_N);
    float l1 = __builtin_nontemporal_load(Lcol + (size_t)(k0 + 1) * DNC_N);
    v4f wp = *(const v4f*)(pkBB + 2 * k0);  // {1-ww0, ww0, 1-ww1, ww1}
    v2f bv = *(const v2f*)(bpB + (size_t)(k0 >> 1) * 32 + 2 * col);
    v2f a;
    a.x = fmaf(wp.x - wwn, l0, wp.y * pn);
    a.y = fmaf(wp.z - wwn, l1, wp.w * pn);
    if (diag) {
      a.x = (an == k0) ? 0.f : a.x;
      a.y = (an == k0 + 1) ? 0.f : a.y;
    }
    acc = __builtin_amdgcn_wmma_f32_16x16x4_f32(false, a, false, bv, (short)0,
                                                acc, false, false);
  };
  for (int mk = 0; mk < n0; mk += 16) {
#pragma unroll
    for (int u = 0; u < 4; ++u) step(mk + u * 4 + half * 2, false);
  }
#pragma unroll
  for (int u = 0; u < 4; ++u) step(n0 + u * 4 + half * 2, true);
  for (int mk = n0 + 16; mk < DNC_N; mk += 16) {
#pragma unroll
    for (int u = 0; u < 4; ++u) step(mk + u * 4 + half * 2, false);
  }
  if (col < DNC_R) {
#pragma unroll
    for (int v = 0; v < 8; ++v) {
      const int m = v + half * 8;
      bwdw[((size_t)b * DNC_N + n0 + m) * DNC_R + col] = acc[v];
    }
  }
#else
  float acc[8];
#pragma unroll
  for (int v = 0; v < 8; ++v) acc[v] = 0.f;
  for (int mk = 0; mk < DNC_N; mk += 2) {
    const float* wp = pkBB + 2 * mk;
    const float* bp = bpB + (size_t)(mk >> 1) * 32 + 2 * col;
#pragma unroll
    for (int v = 0; v < 8; ++v) {
      const int n = n0 + half * 8 + v;
      const float wn = ww[b * DNC_N + n];
      const float pv = prec[b * DNC_N + n];
      const float* Lc = L + (size_t)b * DNC_N * DNC_N + n;
      float a0 = fmaf(wp[0] - wn, Lc[(size_t)mk * DNC_N], wp[1] * pv);
      float a1 = fmaf(wp[2] - wn, Lc[(size_t)(mk + 1) * DNC_N], wp[3] * pv);
      a0 = (mk == n) ? 0.f : a0;
      a1 = (mk + 1 == n) ? 0.f : a1;
      acc[v] += a0 * bp[0] + a1 * bp[1];
    }
  }
  if (col < DNC_R)
#pragma unroll
    for (int v = 0; v < 8; ++v)
      bwdw[((size_t)b * DNC_N + n0 + half * 8 + v) * DNC_R + col] = acc[v];
#endif
}

// ---------------------------------------------------------------------------
// K8: combine read modes -> final read weighting
// ---------------------------------------------------------------------------
__global__ __launch_bounds__(256) void rwf_kernel(
    const float* __restrict__ rmodes, const float* __restrict__ fwdw,
    const float* __restrict__ bwdw, const float* __restrict__ cbr,
    float* __restrict__ rwf) {
  const int i = blockIdx.x * 256 + threadIdx.x;
  if (i >= DNC_B * DNC_N * DNC_R) return;
  const int b = i / (DNC_N * DNC_R);
  const int r = i & 3;
  rwf[i] = rmodes[b * 12 + 0 + r] * bwdw[i] + rmodes[b * 12 + 4 + r] * cbr[i] +
           rmodes[b * 12 + 8 + r] * fwdw[i];
}

// ---------------------------------------------------------------------------
// K9/K10: read_vectors[b,w,r] = sum_n memn[b,n,w]*rwf[b,n,r]
// deterministic two-stage reduction (no float atomics)
// ---------------------------------------------------------------------------
__global__ __launch_bounds__(256) void readvec_partial_kernel(
    const float* __restrict__ memn, const float* __restrict__ rwf,
    float* __restrict__ part) {
  const int b = blockIdx.y;
  const int c = blockIdx.x;  // n-chunk of 256
  const int t = threadIdx.x;
  const int w = t & 63;
  const int g = t >> 6;  // 0..3
  float a0 = 0.f, a1 = 0.f, a2 = 0.f, a3 = 0.f;
  for (int it = 0; it < 64; ++it) {
    const int n = c * 256 + it * 4 + g;
    const float mv = memn[((size_t)b * DNC_N + n) * DNC_W + w];
    const v4f rp = *(const v4f*)(rwf + ((size_t)b * DNC_N + n) * DNC_R);
    a0 = fmaf(mv, rp.x, a0);
    a1 = fmaf(mv, rp.y, a1);
    a2 = fmaf(mv, rp.z, a2);
    a3 = fmaf(mv, rp.w, a3);
  }
  __shared__ float red[1024];
  red[t * 4 + 0] = a0;
  red[t * 4 + 1] = a1;
  red[t * 4 + 2] = a2;
  red[t * 4 + 3] = a3;
  __syncthreads();
  if (g == 0) {
#pragma unroll
    for (int r = 0; r < 4; ++r) {
      const float s = red[w * 4 + r] + red[(w + 64) * 4 + r] +
                      red[(w + 128) * 4 + r] + red[(w + 192) * 4 + r];
      part[((size_t)c * DNC_B + b) * 256 + w * 4 + r] = s;
    }
  }
}

__global__ __launch_bounds__(256) void readvec_final_kernel(
    const float* __restrict__ part, float* __restrict__ out) {
  const int i = blockIdx.x * 256 + threadIdx.x;
  if (i >= DNC_B * DNC_W * DNC_R) return;
  const int b = i >> 8;
  const int wr = i & 255;
  float s = 0.f;
#pragma unroll
  for (int c = 0; c < 16; ++c) s += part[((size_t)c * DNC_B + b) * 256 + wr];
  out[i] = s;
}

// ---------------------------------------------------------------------------
extern "C" void kernel_launch(void* const* d_in, const int* in_sizes, int n_in,
                              void* d_out, int out_size, void* d_ws,
                              size_t ws_size, hipStream_t stream) {
  const float* erase = (const float*)d_in[0];
  const float* fgates = (const float*)d_in[1];
  const float* agate = (const float*)d_in[2];
  const float* wgate = (const float*)d_in[3];
  const float* rmodes = (const float*)d_in[4];
  const float* rstr = (const float*)d_in[5];
  const float* rkeys = (const float*)d_in[6];
  const float* wvec = (const float*)d_in[7];
  const float* wkey = (const float*)d_in[8];
  const float* wstr = (const float*)d_in[9];
  const float* mem = (const float*)d_in[10];
  const float* rdw = (const float*)d_in[11];
  const float* wwt = (const float*)d_in[12];
  const float* musage = (const float*)d_in[13];
  const float* link = (const float*)d_in[14];
  const float* prec = (const float*)d_in[15];
  float* out = (float*)d_out;

  const int BN = DNC_B * DNC_N;
  float* ws = (float*)d_ws;
  float* usage = ws;                          // BN
  float* ssim = usage + BN;                   // BN
  float* cbw = ssim + BN;                     // BN
  float* wwv = cbw + BN;                      // BN
  float* memn = wwv + BN;                     // BN*W
  float* ssr = memn + (size_t)BN * DNC_W;     // BN*R
  float* cbr = ssr + (size_t)BN * DNC_R;      // BN*R
  float* fwdw = cbr + (size_t)BN * DNC_R;     // BN*R
  float* bwdw = fwdw + (size_t)BN * DNC_R;    // BN*R
  float* rwf = bwdw + (size_t)BN * DNC_R;     // BN*R
  float* normk = rwf + (size_t)BN * DNC_R;    // B
  float* normrk = normk + DNC_B;              // B*R
  float* part = normrk + DNC_B * DNC_R;       // 16*B*256
  float* pkA = part + 16 * DNC_B * 256;       // 2*BN
  float* pkB = pkA + (size_t)2 * BN;          // 2*BN
  float* bpack = pkB + (size_t)2 * BN;        // 16*BN

  hipLaunchKernelGGL(norms_kernel, dim3(1), dim3(64), 0, stream, wkey, rkeys,
                     normk, normrk);
  hipLaunchKernelGGL(usage_ssim_kernel, dim3(BN / 256), dim3(256), 0, stream,
                     mem, wkey, musage, wwt, rdw, fgates, wstr, normk, usage,
                     ssim);
  hipLaunchKernelGGL(softmax_cbw_kernel, dim3(DNC_B), dim3(256), 0, stream,
                     ssim, cbw);
  hipLaunchKernelGGL(sort_alloc_ww_kernel, dim3(DNC_B), dim3(1024), 0, stream,
                     usage, cbw, agate, wgate, wwv);
  hipLaunchKernelGGL(prep_kernel, dim3(BN / 256), dim3(256), 0, stream, wwv,
                     prec, rdw, pkA, pkB, bpack);
  hipLaunchKernelGGL(memupdate_kernel, dim3(BN / 256), dim3(256), 0, stream,
                     mem, wwv, erase, wvec, rkeys, rstr, normrk, memn, ssr);
  hipLaunchKernelGGL(softmax_cbr_kernel, dim3(DNC_B * DNC_R), dim3(256), 0,
                     stream, ssr, cbr);
  hipLaunchKernelGGL(link_fwd_kernel, dim3(DNC_N / 128, DNC_B), dim3(256), 0,
                     stream, link, pkA, bpack, wwv, fwdw);
  hipLaunchKernelGGL(link_bwd_kernel, dim3(DNC_N / 128, DNC_B), dim3(256), 0,
                     stream, link, pkB, bpack, wwv, prec, bwdw);
  hipLaunchKernelGGL(rwf_kernel, dim3((BN * DNC_R) / 256), dim3(256), 0,
                     stream, rmodes, fwdw, bwdw, cbr, rwf);
  hipLaunchKernelGGL(readvec_partial_kernel, dim3(16, DNC_B), dim3(256), 0,
                     stream, memn, rwf, part);
  hipLaunchKernelGGL(readvec_final_kernel,
                     dim3((DNC_B * DNC_W * DNC_R) / 256), dim3(256), 0, stream,
                     part, out);
}